// FourierNeuralOperator3D_34789235098210
// MI455X (gfx1250) — compile-verified
//
#include <hip/hip_runtime.h>
#include <hip/hip_bf16.h>
#include <math.h>

// ---------------------------------------------------------------------------
// FNO3D for MI455X (gfx1250, wave32, WMMA).
// FFTs replaced by partial direct DFTs over the 24x24x12 retained modes.
// fp32 WMMA (v_wmma_f32_16x16x4_f32) used for the dense GEMM stages.
// Spectral-weight contraction (680MB stream, the bandwidth floor) uses b128
// non-temporal loads + global prefetch.
// ---------------------------------------------------------------------------

typedef float v2f __attribute__((ext_vector_type(2)));
typedef float v4f __attribute__((ext_vector_type(4)));
typedef float v8f __attribute__((ext_vector_type(8)));

#define NP 262144  // 64^3 spatial points

// ---- workspace layout (float offsets), total ~43.4M floats (~174 MB) ------
#define OFF_V     ((size_t)0)                       // v[p][c]        16.78M
#define OFF_B     ((size_t)16777216)                // R / y / s      16.78M
#define OFF_A1    ((size_t)(16777216*2))            // A1 / A6         6.29M
#define OFF_A2    (OFF_A1 + (size_t)6291456)        // A2 / A5         2.36M
#define OFF_O     (OFF_A2 + (size_t)2359296)        // mode out        0.88M
#define OFF_TAB   (OFF_O  + (size_t)884736)
#define OFF_BAS1  (OFF_TAB)                         // 32x64 fwd-W basis
#define OFF_C24   (OFF_TAB + 2048)                  // cos[24 freq][64]
#define OFF_S24   (OFF_TAB + 3584)                  // sin[24 freq][64]
#define OFF_ICW   (OFF_TAB + 5120)                  // inv-W cos (scaled)
#define OFF_ISW   (OFF_TAB + 5888)                  // inv-W -sin (scaled)
#define OFF_FH    (OFF_TAB + 8192)                  // film hidden 1024
#define OFF_FILM  (OFF_FH + 1024)                   // film 384
#define OFF_STATS (OFF_FILM + 512)                  // GN mean/istd per group
#define OFF_PART  (OFF_STATS + 256)                 // GN partials 2048*128

__device__ __forceinline__ float ang64(int t) {
  return 0.09817477042468103f * (float)t;  // 2*pi/64
}
__device__ __forceinline__ float gelu_f(float x) {
  return 0.5f * x * (1.0f + erff(x * 0.70710678118654752f));
}
__device__ __forceinline__ v8f wmma4(v2f a, v2f b, v8f c) {
  // D = A(16x4 f32) * B(4x16 f32) + C(16x16 f32)
  return __builtin_amdgcn_wmma_f32_16x16x4_f32(false, a, false, b, (short)0, c,
                                               false, false);
}

// ---------------------------------------------------------------------------
// Basis tables (recomputed each launch; ws is poisoned between runs).
__global__ void k_tables(float* ws) {
  int t = threadIdx.x;
  for (int i = t; i < 2048; i += 256) {  // fwd W basis, padded to 32 rows
    int r = i >> 6, w = i & 63;
    float v = 0.f;
    if (r < 12) v = cosf(ang64((r * w) & 63));
    else if (r < 24) v = -sinf(ang64(((r - 12) * w) & 63));
    ws[OFF_BAS1 + i] = v;
  }
  for (int i = t; i < 1536; i += 256) {  // 24-freq set {0..11, 52..63}
    int k = i >> 6, h = i & 63;
    int kf = (k < 12) ? k : k + 40;
    float th = ang64((kf * h) & 63);
    ws[OFF_C24 + i] = cosf(th);
    ws[OFF_S24 + i] = sinf(th);
  }
  for (int i = t; i < 768; i += 256) {   // hermitian inverse-W, 1/64^3 folded
    int kz = i >> 6, w = i & 63;
    float m = ((kz == 0) ? 1.f : 2.f) / 262144.f;
    float th = ang64((kz * w) & 63);
    ws[OFF_ICW + i] = m * cosf(th);
    ws[OFF_ISW + i] = -m * sinf(th);
  }
}

// ---------------------------------------------------------------------------
// FiLM conditioner MLP (tiny).
__global__ void k_film1(const float* lat, const float* w, const float* b,
                        float* ws) {
  int t = blockIdx.x * 256 + threadIdx.x;
  if (t >= 1024) return;
  float a = b[t];
  const float* wr = w + (size_t)t * 512;
  for (int k = 0; k < 512; ++k) a += lat[k] * wr[k];
  ws[OFF_FH + t] = gelu_f(a);
}
__global__ void k_film2(const float* w, const float* b, float* ws) {
  int t = blockIdx.x * 256 + threadIdx.x;
  if (t >= 384) return;
  float a = b[t];
  const float* wr = w + (size_t)t * 1024;
  const float* h = ws + OFF_FH;
  for (int k = 0; k < 1024; ++k) a += h[k] * wr[k];
  ws[OFF_FILM + t] = a;  // t = l*128 + c*2 + phase
}

// ---------------------------------------------------------------------------
// Lift 4->64 channels, write channel-fast v[p][64].
__global__ void k_lift(const float* x, const float* w, const float* b,
                       float* ws) {
  int t = blockIdx.x * 256 + threadIdx.x;
  int c = t & 63, p = t >> 6;
  float a = b[c];
#pragma unroll
  for (int i = 0; i < 4; ++i) a += w[c * 4 + i] * x[i * NP + p];
  ws[OFF_V + (size_t)t] = a;
}

// ---------------------------------------------------------------------------
// S1: partial DFT along W. Per (d,h): out(32pad x 64c) = Bas(32x64w)*V(64w x 64c)
// WMMA f32 16x16x4, 8 waves = 8 output tiles.
__global__ void k_dftw(float* ws) {
  __shared__ float lV[4096];
  __shared__ float lB[2048];
  int dh = blockIdx.x, t = threadIdx.x;
  const float* v = ws + OFF_V + (size_t)dh * 4096;
  for (int i = t; i < 4096; i += 256) lV[i] = v[i];
  for (int i = t; i < 2048; i += 256) lB[i] = ws[OFF_BAS1 + i];
  __syncthreads();
  int wv = t >> 5, l = t & 31;
  int tm = wv >> 2, tn = wv & 3;
  int m = tm * 16 + (l & 15);
  int n = tn * 16 + (l & 15);
  int hk = (l >> 4) << 1;  // lanes 16-31 carry K+2,K+3
  v8f acc = {0.f, 0.f, 0.f, 0.f, 0.f, 0.f, 0.f, 0.f};
#pragma unroll
  for (int k0 = 0; k0 < 64; k0 += 4) {
    int kk = k0 + hk;
    v2f a;  a.x = lB[m * 64 + kk];  a.y = lB[m * 64 + kk + 1];
    v2f bb; bb.x = lV[kk * 64 + n]; bb.y = lV[(kk + 1) * 64 + n];
    acc = wmma4(a, bb, acc);
  }
  int mbase = tm * 16 + ((l >> 4) << 3);
  float* o = ws + OFF_A1 + (size_t)dh * 24 * 64;
#pragma unroll
  for (int r = 0; r < 8; ++r) {
    int mr = mbase + r;
    if (mr < 24) o[mr * 64 + n] = acc[r];  // rows 0..11 re, 12..23 im
  }
}

// S2: DFT along H (complex). A1[d][h][24][c] -> A2[d][ky24][ph2][kz12][c]
__global__ void k_dfth(float* ws) {
  __shared__ float lC[1536], lS[1536];
  int t0 = threadIdx.x;
  for (int i = t0; i < 1536; i += 256) { lC[i] = ws[OFF_C24 + i]; lS[i] = ws[OFF_S24 + i]; }
  __syncthreads();
  int t = blockIdx.x * 256 + t0;
  int c = t & 63, kz = (t >> 6) % 12, ky = (t / 768) % 24, d = t / 18432;
  const float* A1 = ws + OFF_A1;
  float re = 0.f, im = 0.f;
  for (int h = 0; h < 64; ++h) {
    float a = A1[((size_t)(d * 64 + h) * 24 + kz) * 64 + c];
    float b = A1[((size_t)(d * 64 + h) * 24 + 12 + kz) * 64 + c];
    float ct = lC[ky * 64 + h], st = lS[ky * 64 + h];
    re += ct * a + st * b;   // (c - i s)(a + i b)
    im += ct * b - st * a;
  }
  float* A2 = ws + OFF_A2;
  A2[(((size_t)(d * 24 + ky) * 2 + 0) * 12 + kz) * 64 + c] = re;
  A2[(((size_t)(d * 24 + ky) * 2 + 1) * 12 + kz) * 64 + c] = im;
}

// S3: DFT along D -> region-major R[region][c][mode][{re,im}] at OFF_B.
__global__ void k_dftd(float* ws) {
  __shared__ float lC[1536], lS[1536];
  int t0 = threadIdx.x;
  for (int i = t0; i < 1536; i += 256) { lC[i] = ws[OFF_C24 + i]; lS[i] = ws[OFF_S24 + i]; }
  __syncthreads();
  int t = blockIdx.x * 256 + t0;  // 442368 threads
  int kz = t % 12, kyr = (t / 12) % 12, kxr = (t / 144) % 12;
  int reg = (t / 1728) & 3, c = t / 6912;
  int kx = kxr + ((reg & 1) ? 12 : 0);
  int ky = kyr + ((reg >> 1) ? 12 : 0);
  const float* A2 = ws + OFF_A2;
  float re = 0.f, im = 0.f;
  for (int d = 0; d < 64; ++d) {
    float a = A2[(((size_t)(d * 24 + ky) * 2 + 0) * 12 + kz) * 64 + c];
    float b = A2[(((size_t)(d * 24 + ky) * 2 + 1) * 12 + kz) * 64 + c];
    float ct = lC[kx * 64 + d], st = lS[kx * 64 + d];
    re += ct * a + st * b;
    im += ct * b - st * a;
  }
  int mode = (kxr * 12 + kyr) * 12 + kz;
  float* R = ws + OFF_B;
  size_t q = ((size_t)(reg * 64 + c) * 1728 + mode) * 2;
  R[q] = re; R[q + 1] = im;
}

// ---------------------------------------------------------------------------
// Spectral mode contraction: out[o,mode] = sum_i in[i,mode]*w[i,o,mode]
// (complex). This streams the 680MB of weights -> the bandwidth floor.
// One block per (region,o); each lane owns 3 pairs of consecutive modes so
// weight/input loads are contiguous, 16B-aligned b128 across the wave.
// Weights are use-once: non-temporal loads keep L2 for the reused R buffer;
// global prefetch runs 2 i-iterations ahead of the stream.
__global__ void __launch_bounds__(288) k_modes(const float* w1, const float* w2,
                                               const float* w3, const float* w4,
                                               int layer, float* ws) {
  int reg = blockIdx.x >> 6, o = blockIdx.x & 63, t = threadIdx.x;
  const float* wt =
      (reg == 0 ? w1 : reg == 1 ? w2 : reg == 2 ? w3 : w4) +
      (size_t)layer * 14155776;  // 64*64*1728*2
  const float* R = ws + OFF_B;
  float re[6] = {0, 0, 0, 0, 0, 0}, im[6] = {0, 0, 0, 0, 0, 0};
  for (int i = 0; i < 64; ++i) {
    const v4f* wp = (const v4f*)(wt + (size_t)(i * 64 + o) * 3456);
    const v4f* ip = (const v4f*)(R + (size_t)(reg * 64 + i) * 3456);
    if (i + 2 < 64) {  // prefetch the weight stream 2 iterations ahead
      const char* nw = (const char*)(wt + (size_t)((i + 2) * 64 + o) * 3456);
      __builtin_prefetch(nw + t * 48, 0, 0);  // 288*48B covers the 13824B block
    }
#pragma unroll
    for (int j = 0; j < 3; ++j) {
      int mp = j * 288 + t;  // pair of modes {2mp, 2mp+1}
      v4f wv = __builtin_nontemporal_load(wp + mp);
      v4f iv = ip[mp];
      re[2 * j]     += wv.x * iv.x - wv.y * iv.y;
      im[2 * j]     += wv.x * iv.y + wv.y * iv.x;
      re[2 * j + 1] += wv.z * iv.z - wv.w * iv.w;
      im[2 * j + 1] += wv.z * iv.w + wv.w * iv.z;
    }
  }
  v4f* O = (v4f*)(ws + OFF_O + (size_t)(reg * 64 + o) * 3456);
#pragma unroll
  for (int j = 0; j < 3; ++j) {
    int mp = j * 288 + t;
    v4f ov;
    ov.x = re[2 * j];     ov.y = im[2 * j];
    ov.z = re[2 * j + 1]; ov.w = im[2 * j + 1];
    O[mp] = ov;
  }
}

// ---------------------------------------------------------------------------
// S5: inverse DFT along D: O -> A5[d][ky][ph][kz][c] (at OFF_A2).
__global__ void k_idftd(float* ws) {
  __shared__ float lC[1536], lS[1536];
  int t0 = threadIdx.x;
  for (int i = t0; i < 1536; i += 256) { lC[i] = ws[OFF_C24 + i]; lS[i] = ws[OFF_S24 + i]; }
  __syncthreads();
  int t = blockIdx.x * 256 + t0;
  int c = t & 63, kz = (t >> 6) % 12, ky = (t / 768) % 24, d = t / 18432;
  const float* O = ws + OFF_O;
  float re = 0.f, im = 0.f;
  for (int kx = 0; kx < 24; ++kx) {
    int reg = (kx >= 12 ? 1 : 0) + (ky >= 12 ? 2 : 0);
    int mode = ((kx % 12) * 12 + (ky % 12)) * 12 + kz;
    size_t q = ((size_t)(reg * 64 + c) * 1728 + mode) * 2;
    float a = O[q], b = O[q + 1];
    float ct = lC[kx * 64 + d], st = lS[kx * 64 + d];
    re += ct * a - st * b;   // (c + i s)(a + i b)
    im += ct * b + st * a;
  }
  float* A5 = ws + OFF_A2;
  A5[(((size_t)(d * 24 + ky) * 2 + 0) * 12 + kz) * 64 + c] = re;
  A5[(((size_t)(d * 24 + ky) * 2 + 1) * 12 + kz) * 64 + c] = im;
}

// S6: inverse DFT along H: A5 -> A6[d][h][ph][kz][c] (at OFF_A1).
__global__ void k_idfth(float* ws) {
  __shared__ float lC[1536], lS[1536];
  int t0 = threadIdx.x;
  for (int i = t0; i < 1536; i += 256) { lC[i] = ws[OFF_C24 + i]; lS[i] = ws[OFF_S24 + i]; }
  __syncthreads();
  int t = blockIdx.x * 256 + t0;
  int c = t & 63, kz = (t >> 6) % 12, h = (t / 768) % 64, d = t / 49152;
  const float* A5 = ws + OFF_A2;
  float re = 0.f, im = 0.f;
  for (int ky = 0; ky < 24; ++ky) {
    float a = A5[(((size_t)(d * 24 + ky) * 2 + 0) * 12 + kz) * 64 + c];
    float b = A5[(((size_t)(d * 24 + ky) * 2 + 1) * 12 + kz) * 64 + c];
    float ct = lC[ky * 64 + h], st = lS[ky * 64 + h];
    re += ct * a - st * b;
    im += ct * b + st * a;
  }
  float* A6 = ws + OFF_A1;
  A6[(((size_t)(d * 64 + h) * 2 + 0) * 12 + kz) * 64 + c] = re;
  A6[(((size_t)(d * 64 + h) * 2 + 1) * 12 + kz) * 64 + c] = im;
}

// S7: hermitian inverse along W (normalization folded): y[p][c] at OFF_B.
__global__ void k_idftw(float* ws) {
  __shared__ float lic[768], lis[768];
  int t0 = threadIdx.x;
  for (int i = t0; i < 768; i += 256) { lic[i] = ws[OFF_ICW + i]; lis[i] = ws[OFF_ISW + i]; }
  __syncthreads();
  int t = blockIdx.x * 256 + t0;
  int c = t & 63, w = (t >> 6) & 63, h = (t >> 12) & 63, d = t >> 18;
  const float* A6 = ws + OFF_A1;
  float acc = 0.f;
#pragma unroll
  for (int kz = 0; kz < 12; ++kz) {
    float a = A6[(((size_t)(d * 64 + h) * 2 + 0) * 12 + kz) * 64 + c];
    float b = A6[(((size_t)(d * 64 + h) * 2 + 1) * 12 + kz) * 64 + c];
    acc += lic[kz * 64 + w] * a + lis[kz * 64 + w] * b;
  }
  ws[OFF_B + (size_t)t] = acc;
}

// ---------------------------------------------------------------------------
// Pointwise conv (WMMA) fused with residual add + deterministic GN partials.
// s[p][c] = y[p][c] + sum_i conv_w[o,i] v[p][i] + b[c]; per-block channel
// partial (sum, sumsq) written to OFF_PART in fixed slots (no atomics).
__global__ void k_conv_gn(const float* cw, const float* cb, int layer,
                          float* ws) {
  __shared__ float lA[8192];   // 128 p x 64 c
  __shared__ float lBw[4096];  // B[k=i][n=o]
  __shared__ float ps[1024], pq[1024];  // [n=64][slot=16]
  int t = threadIdx.x, p0 = blockIdx.x * 128;
  const float* v = ws + OFF_V + (size_t)p0 * 64;
  for (int i = t; i < 8192; i += 256) lA[i] = v[i];
  const float* wsrc = cw + (size_t)layer * 4096;
  for (int q = t; q < 4096; q += 256) {
    int i = q >> 6, o = q & 63;
    lBw[q] = wsrc[o * 64 + i];  // transpose
  }
  __syncthreads();
  int wv = t >> 5, l = t & 31;
  int m = wv * 16 + (l & 15);
  int hk = (l >> 4) << 1;
  v8f z = {0.f, 0.f, 0.f, 0.f, 0.f, 0.f, 0.f, 0.f};
  v8f acc[4] = {z, z, z, z};
#pragma unroll
  for (int k0 = 0; k0 < 64; k0 += 4) {
    int kk = k0 + hk;
    v2f a; a.x = lA[m * 64 + kk]; a.y = lA[m * 64 + kk + 1];
#pragma unroll
    for (int tn = 0; tn < 4; ++tn) {
      int n = tn * 16 + (l & 15);
      v2f bb; bb.x = lBw[kk * 64 + n]; bb.y = lBw[(kk + 1) * 64 + n];
      acc[tn] = wmma4(a, bb, acc[tn]);
    }
  }
  float* y = ws + OFF_B;
  int mbase = wv * 16 + ((l >> 4) << 3);
  int slot = wv * 2 + (l >> 4);
#pragma unroll
  for (int tn = 0; tn < 4; ++tn) {
    int n = tn * 16 + (l & 15);
    float bias = cb[layer * 64 + n];
    float lsum = 0.f, lsq = 0.f;
#pragma unroll
    for (int r = 0; r < 8; ++r) {
      size_t ad = (size_t)(p0 + mbase + r) * 64 + n;
      float s = acc[tn][r] + bias + y[ad];
      y[ad] = s;  // s overwrites y in-place
      lsum += s; lsq += s * s;
    }
    ps[n * 16 + slot] = lsum;
    pq[n * 16 + slot] = lsq;
  }
  __syncthreads();
  if (t < 64) {
    float s1 = 0.f, s2 = 0.f;
    for (int k = 0; k < 16; ++k) { s1 += ps[t * 16 + k]; s2 += pq[t * 16 + k]; }
    float* part = ws + OFF_PART;
    part[(size_t)blockIdx.x * 128 + t] = s1;
    part[(size_t)blockIdx.x * 128 + 64 + t] = s2;
  }
}

// Deterministic GN stat reduction: 8 groups of 8 channels over 2048 blocks.
__global__ void k_gn_final(float* ws) {
  __shared__ float st[128];
  int t = threadIdx.x;
  const float* part = ws + OFF_PART;
  float s = 0.f;
  int c = t & 63, which = t >> 6;  // 0: sum, 1: sumsq
  for (int b = 0; b < 2048; ++b) s += part[(size_t)b * 128 + which * 64 + c];
  st[t] = s;
  __syncthreads();
  if (t < 8) {
    float sm = 0.f, sq = 0.f;
    for (int cc = 0; cc < 8; ++cc) { sm += st[t * 8 + cc]; sq += st[64 + t * 8 + cc]; }
    float inv = 1.f / 2097152.f;  // 8 ch * 64^3
    float mean = sm * inv;
    float var = sq * inv - mean * mean;
    ws[OFF_STATS + t * 2] = mean;
    ws[OFF_STATS + t * 2 + 1] = rsqrtf(var + 1e-5f);
  }
}

// GN affine + exact GELU + FiLM -> next v.
__global__ void k_gn_apply(const float* gg, const float* gb, int layer,
                           float* ws) {
  int t = blockIdx.x * 256 + threadIdx.x;
  int c = t & 63, g = c >> 3;
  float mean = ws[OFF_STATS + g * 2], istd = ws[OFF_STATS + g * 2 + 1];
  float s = ws[OFF_B + (size_t)t];
  float xn = (s - mean) * istd * gg[layer * 64 + c] + gb[layer * 64 + c];
  float ge = gelu_f(xn);
  float sc = ws[OFF_FILM + layer * 128 + c * 2];
  float sh = ws[OFF_FILM + layer * 128 + c * 2 + 1];
  ws[OFF_V + (size_t)t] = ge * (1.f + sc) + sh;
}

// ---------------------------------------------------------------------------
// Fused projection: h = gelu(v @ p1^T + b1) via WMMA (h staged in LDS),
// out = h @ p2^T + b2 (N=3, VALU). Block: 64 points, 4 waves.
__global__ void __launch_bounds__(128) k_proj(const float* w1, const float* b1,
                                              const float* w2, const float* b2,
                                              float* out, float* ws) {
  __shared__ float lA[4096];   // 64 p x 64 c
  __shared__ float lH[8192];   // 64 p x 128 h
  __shared__ float lw2[384];
  __shared__ float lb1[128];
  int t = threadIdx.x, p0 = blockIdx.x * 64;
  const float* v = ws + OFF_V + (size_t)p0 * 64;
  for (int i = t; i < 4096; i += 128) lA[i] = v[i];
  for (int q = t; q < 384; q += 128) lw2[q] = w2[q];
  if (t < 128) lb1[t] = b1[t];
  __syncthreads();
  int wv = t >> 5, l = t & 31;
  int m = wv * 16 + (l & 15);
  int hk = (l >> 4) << 1;
  v8f z = {0.f, 0.f, 0.f, 0.f, 0.f, 0.f, 0.f, 0.f};
  v8f acc[8] = {z, z, z, z, z, z, z, z};
#pragma unroll
  for (int k0 = 0; k0 < 64; k0 += 4) {
    int kk = k0 + hk;
    v2f a; a.x = lA[m * 64 + kk]; a.y = lA[m * 64 + kk + 1];
#pragma unroll
    for (int tn = 0; tn < 8; ++tn) {
      int n = tn * 16 + (l & 15);
      // B[k][n] = p1_w[n][k]; p1_w is 32KB -> stays hot in cache
      v2f bb; bb.x = w1[n * 64 + kk]; bb.y = w1[n * 64 + kk + 1];
      acc[tn] = wmma4(a, bb, acc[tn]);
    }
  }
  int mbase = wv * 16 + ((l >> 4) << 3);
#pragma unroll
  for (int tn = 0; tn < 8; ++tn) {
    int n = tn * 16 + (l & 15);
#pragma unroll
    for (int r = 0; r < 8; ++r)
      lH[(mbase + r) * 128 + n] = gelu_f(acc[tn][r] + lb1[n]);
  }
  __syncthreads();
  for (int q = t; q < 192; q += 128) {  // 64 p x 3 out
    int pl = q & 63, o = q >> 6;
    float a2 = b2[o];
    for (int j = 0; j < 128; ++j) a2 += lH[pl * 128 + j] * lw2[o * 128 + j];
    out[(size_t)o * NP + p0 + pl] = a2;
  }
}

// ---------------------------------------------------------------------------
extern "C" void kernel_launch(void* const* d_in, const int* in_sizes, int n_in,
                              void* d_out, int out_size, void* d_ws,
                              size_t ws_size, hipStream_t stream) {
  (void)in_sizes; (void)n_in; (void)out_size; (void)ws_size;
  const float* x      = (const float*)d_in[0];
  const float* latent = (const float*)d_in[1];
  const float* lift_w = (const float*)d_in[2];
  const float* lift_b = (const float*)d_in[3];
  const float* sc_w1  = (const float*)d_in[4];
  const float* sc_w2  = (const float*)d_in[5];
  const float* sc_w3  = (const float*)d_in[6];
  const float* sc_w4  = (const float*)d_in[7];
  const float* conv_w = (const float*)d_in[8];
  const float* conv_b = (const float*)d_in[9];
  const float* gn_g   = (const float*)d_in[10];
  const float* gn_b   = (const float*)d_in[11];
  const float* p1_w   = (const float*)d_in[12];
  const float* p1_b   = (const float*)d_in[13];
  const float* p2_w   = (const float*)d_in[14];
  const float* p2_b   = (const float*)d_in[15];
  const float* c1_w   = (const float*)d_in[16];
  const float* c1_b   = (const float*)d_in[17];
  const float* c2_w   = (const float*)d_in[18];
  const float* c2_b   = (const float*)d_in[19];
  float* ws  = (float*)d_ws;
  float* out = (float*)d_out;

  k_tables<<<1, 256, 0, stream>>>(ws);
  k_film1<<<4, 256, 0, stream>>>(latent, c1_w, c1_b, ws);
  k_film2<<<2, 256, 0, stream>>>(c2_w, c2_b, ws);
  k_lift<<<65536, 256, 0, stream>>>(x, lift_w, lift_b, ws);

  for (int l = 0; l < 3; ++l) {
    k_dftw<<<4096, 256, 0, stream>>>(ws);
    k_dfth<<<4608, 256, 0, stream>>>(ws);
    k_dftd<<<1728, 256, 0, stream>>>(ws);
    k_modes<<<256, 288, 0, stream>>>(sc_w1, sc_w2, sc_w3, sc_w4, l, ws);
    k_idftd<<<4608, 256, 0, stream>>>(ws);
    k_idfth<<<12288, 256, 0, stream>>>(ws);
    k_idftw<<<65536, 256, 0, stream>>>(ws);
    k_conv_gn<<<2048, 256, 0, stream>>>(conv_w, conv_b, l, ws);
    k_gn_final<<<1, 128, 0, stream>>>(ws);
    k_gn_apply<<<65536, 256, 0, stream>>>(gn_g, gn_b, l, ws);
  }
  k_proj<<<4096, 128, 0, stream>>>(p1_w, p1_b, p2_w, p2_b, out, ws);
}